// SNNLayer_67611375174157
// MI455X (gfx1250) — compile-verified
//
#include <hip/hip_runtime.h>
#include <hip/hip_bf16.h>

// ---------------------------------------------------------------------------
// SNN layer step for MI455X (gfx1250, wave32, WMMA, LDS-staged A operand).
//
// Shapes: B=256, N_IN=4096, N_OUT=4096, M=16 (4-bit membrane), T_MASK=15.
// Output (flat float concat): active_next (1M), mem_fict_new (1M),
// mem_phys_new (1M), lut_new (64K), t_last_new (256).
// ---------------------------------------------------------------------------

#define B_DIM    256
#define N_IN     4096
#define N_OUT    4096
#define M_MASK   15
#define T_MASK   15
#define ALPHA    0.5f

#define KCHUNK   256          // K-extent staged in LDS per phase
#define LDSW     260          // padded row stride (floats): 4*lane mod 64 banks
                              // -> conflict-free ds_load_b64 across 16 lanes

typedef float v2f __attribute__((ext_vector_type(2)));
typedef float v8f __attribute__((ext_vector_type(8)));

// ---------------------------------------------------------------------------
// K0: initialize the LUT output region with accumulate_lut_delta (input).
// ---------------------------------------------------------------------------
__global__ void snn_lut_init(const float* __restrict__ lut_in,
                             float* __restrict__ lut_out) {
    int i = blockIdx.x * blockDim.x + threadIdx.x;   // 65536 elements
    if (i < N_OUT * 16) lut_out[i] = lut_in[i];
}

// ---------------------------------------------------------------------------
// K1: per-batch scalars. One wave32 per batch row.
//   acc[b]     = XOR over active inputs of prev_neuron_id (low 4 bits)
//   delta_t[b] = (t - t_last[b]) & 15 ; t_last_new[b] = t
// ---------------------------------------------------------------------------
__global__ void snn_rowscan(const int* __restrict__ active_mask,
                            const int* __restrict__ prev_id,
                            const int* __restrict__ t_last,
                            const int* __restrict__ t_scalar,
                            int*       __restrict__ bscal,      // [B][2] = {dt, acc}
                            float*     __restrict__ out_tlast) {
    const int lane = threadIdx.x & 31;
    const int b    = (blockIdx.x * blockDim.x + threadIdx.x) >> 5;  // wave id
    if (b >= B_DIM) return;

    const int* row = active_mask + (size_t)b * N_IN;
    int x = 0;
    #pragma unroll 4
    for (int i = lane; i < N_IN; i += 32)
        x ^= row[i] ? prev_id[i] : 0;

    // wave32 XOR butterfly (5 steps; warpSize == 32 on gfx1250)
    #pragma unroll
    for (int off = 16; off > 0; off >>= 1)
        x ^= __shfl_xor(x, off, 32);

    if (lane == 0) {
        const int t  = t_scalar[0];
        const int dt = (t - t_last[b]) & T_MASK;
        bscal[2 * b + 0] = dt;
        bscal[2 * b + 1] = x & M_MASK;
        out_tlast[b]     = (float)t;
    }
}

// ---------------------------------------------------------------------------
// K2: fp32 WMMA masked matmul + fused epilogue, A staged through LDS.
//
// Block = 256 threads = 8 wave32s. Block covers one 16-row batch tile and
// 8 consecutive 16-col output tiles (one per wave). Grid = 16 x 32 = 512.
//
// Per K-chunk of 256:
//   - 256 threads cooperatively load the 16x256 mask chunk with coalesced
//     b128 loads, convert to f32 once, store to padded LDS.
//   - each wave runs 64 x V_WMMA_F32_16X16X4_F32, A from ds_load_b64
//     (bank-conflict-free), B from 64B-coalesced global loads (L2-resident).
//
// V_WMMA_F32_16X16X4_F32 operand layout (ISA 7.12.2):
//   A (16x4): lane M = lane&15; VGPR0 K = 2*(lane>=16), VGPR1 K+1.
//   B (4x16): VGPR0: K = 2*(lane>=16), N = lane&15; VGPR1: K+1.
//   C (16x16): VGPR r -> row r + 8*(lane>=16), col lane&15.
// ---------------------------------------------------------------------------
__global__ __launch_bounds__(256)
void snn_wmma_fused(const int*   __restrict__ active_mask,  // (B, N_IN)
                    const float* __restrict__ syn_w,        // (N_IN, N_OUT)
                    const float* __restrict__ mem_fict,     // (B, N_OUT)
                    const int*   __restrict__ mem_phys,     // (B, N_OUT)
                    const float* __restrict__ tau,          // (N_OUT)
                    const float* __restrict__ v_th,         // (N_OUT)
                    const float* __restrict__ mem_map,      // (N_OUT, 16)
                    const int*   __restrict__ bscal,        // (B, 2)
                    float* __restrict__ out_active,
                    float* __restrict__ out_fict,
                    float* __restrict__ out_phys,
                    float* __restrict__ out_lut) {
    __shared__ float amask[16 * LDSW];     // 16.6 KB of the 320 KB/WGP pool

    const int tid  = threadIdx.x;
    const int wave = tid >> 5;             // 0..7
    const int lane = tid & 31;             // wave32 lane
    const int half = lane >> 4;            // 0 | 1
    const int l15  = lane & 15;

    const int tb = blockIdx.x & 15;        // 16 batch tiles
    const int tn = blockIdx.x >> 4;        // 32 groups of 128 output cols
    const int b0 = tb << 4;
    const int o0 = (tn << 7) + (wave << 4);  // this wave's 16-col tile

    // cooperative mask-load mapping: 16 consecutive ints per thread, coalesced
    const int lrow  = tid >> 4;            // 0..15
    const int lkoff = (tid & 15) << 4;     // 0,16,...,240
    const int* mrow = active_mask + (size_t)(b0 + lrow) * N_IN + lkoff;
    float*     mdst = &amask[lrow * LDSW + lkoff];

    // this lane's operand bases (K offset 2*half baked in)
    const float* wbase = syn_w + (size_t)(half << 1) * N_OUT + o0 + l15;
    const float* aLds  = &amask[l15 * LDSW + (half << 1)];

    v8f c = {};  // fp32 accumulator, 8 VGPRs

    for (int kc = 0; kc < N_IN; kc += KCHUNK) {
        // ---- stage 16 x KCHUNK mask chunk into LDS as f32 ------------------
        {
            const int4* src = (const int4*)(mrow + kc);
            #pragma unroll
            for (int j = 0; j < 4; ++j) {
                const int4 m = src[j];
                mdst[4 * j + 0] = (float)m.x;
                mdst[4 * j + 1] = (float)m.y;
                mdst[4 * j + 2] = (float)m.z;
                mdst[4 * j + 3] = (float)m.w;
            }
            if (kc + KCHUNK < N_IN)               // global_prefetch_b8
                __builtin_prefetch(mrow + kc + KCHUNK, 0, 1);
        }
        __syncthreads();

        // ---- 64 fp32 WMMAs over this chunk ---------------------------------
        const float* wk = wbase + (size_t)kc * N_OUT;
        #pragma unroll 4
        for (int k = 0; k < KCHUNK; k += 4) {
            v2f a;                                  // ds_load_b64, no conflicts
            a.x = aLds[k];
            a.y = aLds[k + 1];

            v2f bm;                                 // 64B-coalesced per half-wave
            bm.x = wk[(size_t)k * N_OUT];
            bm.y = wk[(size_t)(k + 1) * N_OUT];

            c = __builtin_amdgcn_wmma_f32_16x16x4_f32(
                    /*neg_a=*/false, a, /*neg_b=*/false, bm,
                    /*c_mod=*/(short)0, c, /*reuse_a=*/false, /*reuse_b=*/false);
        }
        __syncthreads();   // protect LDS before next chunk overwrites it
    }

    // ---- fused epilogue: lane owns rows {r, r+8*half} x col l15 ------------
    const int o = o0 + l15;
    const float tau_o = tau[o];
    const float vth_o = v_th[o];

    #pragma unroll
    for (int r = 0; r < 8; ++r) {
        const int b  = b0 + r + (half << 3);
        const int dt = bscal[2 * b + 0];
        const int ac = bscal[2 * b + 1];

        const size_t idx = (size_t)b * N_OUT + o;

        const int   mp = (mem_phys[idx] ^ dt ^ ac) & M_MASK;
        const float v  = mem_map[o * 16 + mp];

        out_active[idx] = (v >= vth_o) ? 1.0f : 0.0f;

        const float decay = __expf(-tau_o * (float)dt);   // v_exp_f32
        out_fict[idx] = mem_fict[idx] * decay + c[r];
        out_phys[idx] = (float)mp;

        atomicAdd(&out_lut[o * 16 + mp], ALPHA * ((float)mp - v));
    }
}

// ---------------------------------------------------------------------------
// Host-side launcher (graph-capture safe: only kernel launches on `stream`).
// ---------------------------------------------------------------------------
extern "C" void kernel_launch(void* const* d_in, const int* in_sizes, int n_in,
                              void* d_out, int out_size, void* d_ws, size_t ws_size,
                              hipStream_t stream) {
    const int*   active_mask = (const int*)  d_in[0];
    const int*   t_scalar    = (const int*)  d_in[1];
    const int*   prev_id     = (const int*)  d_in[2];
    const int*   t_last      = (const int*)  d_in[3];
    const float* mem_fict    = (const float*)d_in[4];
    const int*   mem_phys    = (const int*)  d_in[5];
    const float* lut_in      = (const float*)d_in[6];
    const float* tau         = (const float*)d_in[7];
    const float* v_th        = (const float*)d_in[8];
    const float* syn_w       = (const float*)d_in[9];
    const float* mem_map     = (const float*)d_in[10];

    float* out        = (float*)d_out;
    float* out_active = out;                                  // B*N_OUT
    float* out_fict   = out + (size_t)B_DIM * N_OUT;          // B*N_OUT
    float* out_phys   = out + 2 * (size_t)B_DIM * N_OUT;      // B*N_OUT
    float* out_lut    = out + 3 * (size_t)B_DIM * N_OUT;      // N_OUT*16
    float* out_tlast  = out + 3 * (size_t)B_DIM * N_OUT + (size_t)N_OUT * 16; // B

    int* bscal = (int*)d_ws;  // B * 2 ints of scratch

    // K0: seed LUT output with input accumulate_lut_delta
    snn_lut_init<<<(N_OUT * 16 + 255) / 256, 256, 0, stream>>>(lut_in, out_lut);

    // K1: per-batch parity + delta_t (one wave32 per batch; 8 waves/block)
    snn_rowscan<<<(B_DIM * 32 + 255) / 256, 256, 0, stream>>>(
        active_mask, prev_id, t_last, t_scalar, bscal, out_tlast);

    // K2: WMMA fp32 matmul + fused epilogue
    //     grid = 16 batch-tiles x 32 col-groups, 256 threads (8 wave32s)
    snn_wmma_fused<<<(B_DIM / 16) * (N_OUT / 128), 256, 0, stream>>>(
        active_mask, syn_w, mem_fict, mem_phys, tau, v_th, mem_map, bscal,
        out_active, out_fict, out_phys, out_lut);
}